// GINConvNet_62491774157297
// MI455X (gfx1250) — compile-verified
//
#include <hip/hip_runtime.h>

#define N_NODES 100000
#define N_EDGES 3200000
#define N_FEAT 128
#define DIM 32
#define N_GRAPHS 1024
#define N_CLASSES 10
#define BN_EPS 1e-5f

typedef float v2f __attribute__((ext_vector_type(2)));
typedef float v8f __attribute__((ext_vector_type(8)));

static __device__ __forceinline__ v8f wmma_k4(v2f a, v2f b, v8f c) {
    // D = A(16x4) * B(4x16) + C, all f32  ->  v_wmma_f32_16x16x4_f32
    return __builtin_amdgcn_wmma_f32_16x16x4_f32(false, a, false, b, (short)0, c, false, false);
}

#define V8F_ZERO (v8f){0.f, 0.f, 0.f, 0.f, 0.f, 0.f, 0.f, 0.f}

// ---------------- zero ----------------
__global__ __launch_bounds__(256) void zero_f(float* __restrict__ p, int n) {
    int i = blockIdx.x * 256 + threadIdx.x;
    if (i < n) p[i] = 0.0f;
}

// ---------------- edge scatter-add in 32-dim space ----------------
// 1 wave per edge, 1 feature per lane; node tables are 12.8 MB -> L2-resident
__global__ __launch_bounds__(256) void scatter32(const float* __restrict__ y,
                                                 const int* __restrict__ src,
                                                 const int* __restrict__ dst,
                                                 float* __restrict__ acc) {
    int t = blockIdx.x * 256 + threadIdx.x;
    int e = t >> 5;
    int f = t & 31;
    if (e >= N_EDGES) return;
    int s = src[e], d = dst[e];
    atomicAdd(acc + (size_t)d * DIM + f, y[(size_t)s * DIM + f]);
}

// ---------------- proj1: y = x @ w1a ; s = y (accumulator init fused) ----------------
// block = 128 threads = 4 waves; wave -> 16-node tile; A tile staged in LDS via float4
__global__ __launch_bounds__(128) void proj1(const float* __restrict__ x,
                                             const float* __restrict__ w1a,
                                             float* __restrict__ y,
                                             float* __restrict__ s) {
    __shared__ float sW[N_FEAT * DIM];     // 16 KB
    __shared__ float sA[4][16 * 132];      // 33 KB, ld=132: 16B-aligned rows, conflict-free cols

    const int tid = threadIdx.x;
    for (int i = tid; i < N_FEAT * DIM; i += 128) sW[i] = w1a[i];

    const int wave = tid >> 5, lane = tid & 31;
    const int row0 = (blockIdx.x * 4 + wave) * 16;
    float* __restrict__ T = &sA[wave][0];

    // coalesced float4 stage of the 16x128 A tile (clamped rows)
    for (int p = lane; p < 512; p += 32) {
        int r = p >> 5, c4 = p & 31;
        int gr = row0 + r;
        if (gr >= N_NODES) gr = N_NODES - 1;
        *reinterpret_cast<float4*>(T + r * 132 + c4 * 4) =
            *reinterpret_cast<const float4*>(x + (size_t)gr * N_FEAT + c4 * 4);
    }
    __syncthreads();

    const int colb = lane & 15;
    const int kkb = (lane < 16) ? 0 : 2;
    const int rb = (lane < 16) ? 0 : 8;

    v8f acc0 = V8F_ZERO, acc1 = V8F_ZERO;
    const float* __restrict__ At = T + colb * 132;
    for (int k = 0; k < N_FEAT; k += 4) {
        v2f a = {At[k + kkb], At[k + kkb + 1]};
        v2f b0 = {sW[(k + kkb) * DIM + colb], sW[(k + kkb + 1) * DIM + colb]};
        v2f b1 = {sW[(k + kkb) * DIM + 16 + colb], sW[(k + kkb + 1) * DIM + 16 + colb]};
        acc0 = wmma_k4(a, b0, acc0);
        acc1 = wmma_k4(a, b1, acc1);
    }

    for (int i = 0; i < 8; ++i) {
        int r = row0 + rb + i;
        if (r < N_NODES) {
            size_t o0 = (size_t)r * DIM + colb;
            size_t o1 = (size_t)r * DIM + 16 + colb;
            y[o0] = acc0[i];
            y[o1] = acc1[i];
            s[o0] = acc0[i];   // fused accumulator init (self term of GIN sum)
            s[o1] = acc1[i];
        }
    }
}

// shared 16x16x32 GEMM stage macro (A tile in LDS ld=36, W 32x32 in LDS)
#define GEMM32(W)                                                                  \
    acc0 = V8F_ZERO;                                                               \
    acc1 = V8F_ZERO;                                                               \
    for (int k = 0; k < DIM; k += 4) {                                             \
        v2f a = {At[k + kkb], At[k + kkb + 1]};                                    \
        v2f bf0 = {(W)[(k + kkb) * DIM + colb], (W)[(k + kkb + 1) * DIM + colb]};  \
        v2f bf1 = {(W)[(k + kkb) * DIM + 16 + colb],                               \
                   (W)[(k + kkb + 1) * DIM + 16 + colb]};                          \
        acc0 = wmma_k4(a, bf0, acc0);                                              \
        acc1 = wmma_k4(a, bf1, acc1);                                              \
    }

// ---------------- mlp1 tail + proj2 fused ----------------
// in:  s1 = x@w1a + seg_sum((x@w1a)[src])
// t1 = relu(s1 + b1a); h1 = bn1(relu(t1 @ w1b + b1b)); y2 = h1 @ w2a ; s2 = y2
__global__ __launch_bounds__(256) void mlp1_proj2(const float* __restrict__ s1,
                                                  const float* __restrict__ b1a,
                                                  const float* __restrict__ w1b,
                                                  const float* __restrict__ b1b,
                                                  const float* __restrict__ bg,
                                                  const float* __restrict__ bb,
                                                  const float* __restrict__ bm,
                                                  const float* __restrict__ bv,
                                                  const float* __restrict__ w2a,
                                                  float* __restrict__ y2,
                                                  float* __restrict__ s2) {
    __shared__ float sW0[DIM * DIM], sW1[DIM * DIM];
    __shared__ float sBa[DIM], sBb[DIM], sSc[DIM], sSh[DIM];
    __shared__ float sT[8][16 * 36];

    const int tid = threadIdx.x;
    for (int i = tid; i < DIM * DIM; i += 256) { sW0[i] = w1b[i]; sW1[i] = w2a[i]; }
    if (tid < DIM) {
        sBa[tid] = b1a[tid];
        sBb[tid] = b1b[tid];
        float sc = bg[tid] * rsqrtf(bv[tid] + BN_EPS);
        sSc[tid] = sc;
        sSh[tid] = bb[tid] - bm[tid] * sc;
    }
    __syncthreads();

    const int wave = tid >> 5, lane = tid & 31;
    const int row0 = (blockIdx.x * 8 + wave) * 16;
    const int colb = lane & 15;
    const int kkb = (lane < 16) ? 0 : 2;
    const int rb = (lane < 16) ? 0 : 8;
    float* __restrict__ T = &sT[wave][0];

    // load tile with fused relu(s1 + b1a)
    for (int idx = lane; idx < 16 * DIM; idx += 32) {
        int r = idx >> 5, c = idx & 31;
        int gr = row0 + r;
        if (gr >= N_NODES) gr = N_NODES - 1;
        T[r * 36 + c] = fmaxf(s1[(size_t)gr * DIM + c] + sBa[c], 0.f);
    }
    __syncthreads();

    v8f acc0, acc1;
    const float* __restrict__ At = T + colb * 36;

    // stage 1: h1 = bn1( relu(t1 @ w1b + b1b) )
    GEMM32(sW0);
    __syncthreads();
    for (int i = 0; i < 8; ++i) {
        float v0 = fmaxf(acc0[i] + sBb[colb], 0.f);
        float v1 = fmaxf(acc1[i] + sBb[16 + colb], 0.f);
        T[(rb + i) * 36 + colb]      = v0 * sSc[colb] + sSh[colb];
        T[(rb + i) * 36 + 16 + colb] = v1 * sSc[16 + colb] + sSh[16 + colb];
    }
    __syncthreads();

    // stage 2: y2 = h1 @ w2a  (bias b2a applied after conv2 aggregation)
    GEMM32(sW1);
    for (int i = 0; i < 8; ++i) {
        int r = row0 + rb + i;
        if (r < N_NODES) {
            size_t o0 = (size_t)r * DIM + colb;
            size_t o1 = (size_t)r * DIM + 16 + colb;
            y2[o0] = acc0[i];
            y2[o1] = acc1[i];
            s2[o0] = acc0[i];   // fused accumulator init
            s2[o1] = acc1[i];
        }
    }
}

// ---------------- mlp2 tail + bn2 + fc1 + fc2 fused ----------------
// in: s2 = h1@w2a + seg_sum((h1@w2a)[src])
// t = relu(s2 + b2a); h = bn2(relu(t @ w2b + b2b)); t = relu(h@fc1w+fc1b); hf = t@fc2w+fc2b
__global__ __launch_bounds__(256) void mlp_tail(const float* __restrict__ s2,
                                                const float* __restrict__ b2a,
                                                const float* __restrict__ w2b,
                                                const float* __restrict__ b2b,
                                                const float* __restrict__ bg,
                                                const float* __restrict__ bb,
                                                const float* __restrict__ bm,
                                                const float* __restrict__ bv,
                                                const float* __restrict__ fc1w,
                                                const float* __restrict__ fc1b,
                                                const float* __restrict__ fc2w,
                                                const float* __restrict__ fc2b,
                                                float* __restrict__ hf) {
    __shared__ float sW0[DIM * DIM], sW1[DIM * DIM], sW2[DIM * DIM];
    __shared__ float sBa[DIM], sB0[DIM], sB1[DIM], sB2[DIM], sSc[DIM], sSh[DIM];
    __shared__ float sT[8][16 * 36];

    const int tid = threadIdx.x;
    for (int i = tid; i < DIM * DIM; i += 256) {
        sW0[i] = w2b[i]; sW1[i] = fc1w[i]; sW2[i] = fc2w[i];
    }
    if (tid < DIM) {
        sBa[tid] = b2a[tid];
        sB0[tid] = b2b[tid];
        sB1[tid] = fc1b[tid];
        sB2[tid] = fc2b[tid];
        float sc = bg[tid] * rsqrtf(bv[tid] + BN_EPS);
        sSc[tid] = sc;
        sSh[tid] = bb[tid] - bm[tid] * sc;
    }
    __syncthreads();

    const int wave = tid >> 5, lane = tid & 31;
    const int row0 = (blockIdx.x * 8 + wave) * 16;
    const int colb = lane & 15;
    const int kkb = (lane < 16) ? 0 : 2;
    const int rb = (lane < 16) ? 0 : 8;
    float* __restrict__ T = &sT[wave][0];

    for (int idx = lane; idx < 16 * DIM; idx += 32) {
        int r = idx >> 5, c = idx & 31;
        int gr = row0 + r;
        if (gr >= N_NODES) gr = N_NODES - 1;
        T[r * 36 + c] = fmaxf(s2[(size_t)gr * DIM + c] + sBa[c], 0.f);
    }
    __syncthreads();

    v8f acc0, acc1;
    const float* __restrict__ At = T + colb * 36;

    // stage 1: h = bn2( relu(t @ w2b + b2b) )
    GEMM32(sW0);
    __syncthreads();
    for (int i = 0; i < 8; ++i) {
        float v0 = fmaxf(acc0[i] + sB0[colb], 0.f);
        float v1 = fmaxf(acc1[i] + sB0[16 + colb], 0.f);
        T[(rb + i) * 36 + colb]      = v0 * sSc[colb] + sSh[colb];
        T[(rb + i) * 36 + 16 + colb] = v1 * sSc[16 + colb] + sSh[16 + colb];
    }
    __syncthreads();

    // stage 2: t = relu(h @ fc1w + fc1b)
    GEMM32(sW1);
    __syncthreads();
    for (int i = 0; i < 8; ++i) {
        T[(rb + i) * 36 + colb]      = fmaxf(acc0[i] + sB1[colb], 0.f);
        T[(rb + i) * 36 + 16 + colb] = fmaxf(acc1[i] + sB1[16 + colb], 0.f);
    }
    __syncthreads();

    // stage 3: hf = t @ fc2w + fc2b
    GEMM32(sW2);
    for (int i = 0; i < 8; ++i) {
        int r = row0 + rb + i;
        if (r < N_NODES) {
            hf[(size_t)r * DIM + colb]      = acc0[i] + sB2[colb];
            hf[(size_t)r * DIM + 16 + colb] = acc1[i] + sB2[16 + colb];
        }
    }
}

#undef GEMM32

// ---------------- global mean pool accumulation ----------------
__global__ __launch_bounds__(256) void pool_accum(const float* __restrict__ hf,
                                                  const int* __restrict__ batch,
                                                  float* __restrict__ gsum,
                                                  float* __restrict__ gcnt) {
    int t = blockIdx.x * 256 + threadIdx.x;
    int node = t >> 5;
    int f = t & 31;
    if (node >= N_NODES) return;
    int g = batch[node];
    atomicAdd(gsum + (size_t)g * DIM + f, hf[(size_t)node * DIM + f]);
    if (f == 0) atomicAdd(gcnt + g, 1.0f);
}

// ---------------- final: pooled @ lin_w + lin_b ----------------
__global__ __launch_bounds__(256) void final_lin(const float* __restrict__ gsum,
                                                 const float* __restrict__ gcnt,
                                                 const float* __restrict__ lin_w,
                                                 const float* __restrict__ lin_b,
                                                 float* __restrict__ out) {
    int t = blockIdx.x * 256 + threadIdx.x;
    if (t >= N_GRAPHS * N_CLASSES) return;
    int g = t / N_CLASSES, c = t % N_CLASSES;
    float cnt = gcnt[g];
    cnt = cnt > 1.f ? cnt : 1.f;
    float inv = 1.f / cnt;
    float acc = lin_b[c];
    for (int k = 0; k < DIM; ++k) acc += gsum[g * DIM + k] * inv * lin_w[k * N_CLASSES + c];
    out[t] = acc;
}

extern "C" void kernel_launch(void* const* d_in, const int* in_sizes, int n_in,
                              void* d_out, int out_size, void* d_ws, size_t ws_size,
                              hipStream_t stream) {
    const float* x     = (const float*)d_in[0];
    const int*   ei    = (const int*)d_in[1];
    const int*   batch = (const int*)d_in[2];
    const float* w1a = (const float*)d_in[3];
    const float* b1a = (const float*)d_in[4];
    const float* w1b = (const float*)d_in[5];
    const float* b1b = (const float*)d_in[6];
    const float* bn1g = (const float*)d_in[7];
    const float* bn1b = (const float*)d_in[8];
    const float* bn1m = (const float*)d_in[9];
    const float* bn1v = (const float*)d_in[10];
    const float* w2a = (const float*)d_in[11];
    const float* b2a = (const float*)d_in[12];
    const float* w2b = (const float*)d_in[13];
    const float* b2b = (const float*)d_in[14];
    const float* bn2g = (const float*)d_in[15];
    const float* bn2b = (const float*)d_in[16];
    const float* bn2m = (const float*)d_in[17];
    const float* bn2v = (const float*)d_in[18];
    const float* fc1w = (const float*)d_in[19];
    const float* fc1b = (const float*)d_in[20];
    const float* fc2w = (const float*)d_in[21];
    const float* fc2b = (const float*)d_in[22];
    const float* linw = (const float*)d_in[23];
    const float* linb = (const float*)d_in[24];

    const int* src = ei;
    const int* dst = ei + N_EDGES;

    // two rotating node buffers of N_NODES*DIM floats
    float* ws   = (float*)d_ws;
    float* bufA = ws;                          // 3,200,000 f
    float* bufB = ws + (size_t)N_NODES * DIM;  // 3,200,000 f
    float* gsum = ws + (size_t)2 * N_NODES * DIM;
    float* gcnt = gsum + (size_t)N_GRAPHS * DIM;

    float* out = (float*)d_out;

    const int eThr = N_EDGES * 32;             // 102.4M threads

    // y1 = x @ w1a (WMMA, LDS-staged A); s1 init fused (bufB = y1)
    proj1<<<(N_NODES + 63) / 64, 128, 0, stream>>>(x, w1a, bufA, bufB);

    // s1 += seg_sum(y1[src], dst)   (32-wide edge phase, L2-resident)
    scatter32<<<(eThr + 255) / 256, 256, 0, stream>>>(bufA, src, dst, bufB);

    // h1 = bn1(relu(relu(s1+b1a)@w1b+b1b)); y2 = h1 @ w2a; s2 init fused (bufA = y2 -> copy into... )
    mlp1_proj2<<<(N_NODES + 127) / 128, 256, 0, stream>>>(
        bufB, b1a, w1b, b1b, bn1g, bn1b, bn1m, bn1v, w2a, bufA, bufB);

    // s2 += seg_sum(y2[src], dst)
    scatter32<<<(eThr + 255) / 256, 256, 0, stream>>>(bufA, src, dst, bufB);

    // hf = (bn2(relu(relu(s2+b2a)@w2b+b2b)) -> fc1+relu -> fc2)
    mlp_tail<<<(N_NODES + 127) / 128, 256, 0, stream>>>(
        bufB, b2a, w2b, b2b, bn2g, bn2b, bn2m, bn2v,
        fc1w, fc1b, fc2w, fc2b, bufA);

    // mean pool + classifier
    zero_f<<<(N_GRAPHS * DIM + N_GRAPHS + 255) / 256, 256, 0, stream>>>(
        gsum, N_GRAPHS * DIM + N_GRAPHS);
    pool_accum<<<(N_NODES * 32 + 255) / 256, 256, 0, stream>>>(bufA, batch, gsum, gcnt);
    final_lin<<<(N_GRAPHS * N_CLASSES + 255) / 256, 256, 0, stream>>>(
        gsum, gcnt, linw, linb, out);
}